// MultiHeadTimeDimensionAttention_3685081940408
// MI455X (gfx1250) — compile-verified
//
#include <hip/hip_runtime.h>

#define B_ 16
#define T_ 4096
#define Z_ 1024
#define H_ 16
#define DK_ 64

typedef __attribute__((ext_vector_type(2))) float v2f;
typedef __attribute__((ext_vector_type(4))) float v4f;
typedef __attribute__((ext_vector_type(8))) float v8f;

// ---------------------------------------------------------------------------
// CDNA5 async global->LDS path (ASYNCcnt-tracked, bypasses VGPRs).
// Builtin signature (from hipcc diagnostic): first param is
// `__attribute__((vector_size(16))) int` pointer in the global (as1) space.
// Falls back to synchronous v4f staging if the toolchain lacks the builtins.
// ---------------------------------------------------------------------------
#if defined(__has_builtin)
#if __has_builtin(__builtin_amdgcn_global_load_async_to_lds_b128) && \
    __has_builtin(__builtin_amdgcn_s_wait_asynccnt)
#define HAVE_ASYNC 1
#endif
#endif
#ifndef HAVE_ASYNC
#define HAVE_ASYNC 0
#endif

#if HAVE_ASYNC
typedef int v4i __attribute__((vector_size(16)));
typedef __attribute__((address_space(1))) v4i g_v4i;
typedef __attribute__((address_space(3))) v4i l_v4i;
__device__ __forceinline__ void cp16_async(const float* g, float* l) {
  __builtin_amdgcn_global_load_async_to_lds_b128((g_v4i*)g, (l_v4i*)l, 0, 0);
}
#define ASYNC_WAIT(n) __builtin_amdgcn_s_wait_asynccnt(n)
#else
__device__ __forceinline__ void cp16_async(const float* g, float* l) {
  *(v4f*)l = *(const v4f*)g;
}
#define ASYNC_WAIT(n) ((void)0)
#endif

// ---------------------------------------------------------------------------
// Kernel 1a: q[b,h,d] = o_last[b,:]·Wq[h,:,d] + bq[h,d] ; qb[b,h] = q·bk[h,:]
// ---------------------------------------------------------------------------
__global__ void qproj_kernel(const float* __restrict__ o_last,
                             const float* __restrict__ Wq,
                             const float* __restrict__ bq,
                             const float* __restrict__ bk,
                             float* __restrict__ q,
                             float* __restrict__ qb) {
  int bh = blockIdx.x;
  int b = bh / H_, h = bh % H_;
  int d = threadIdx.x;
  const float* ol = o_last + b * Z_;
  const float* w  = Wq + (size_t)h * Z_ * DK_ + d;
  float acc = bq[h * DK_ + d];
  for (int z = 0; z < Z_; ++z) acc += ol[z] * w[(size_t)z * DK_];
  q[bh * DK_ + d] = acc;
  __shared__ float red[DK_];
  red[d] = acc * bk[h * DK_ + d];
  __syncthreads();
  for (int s = DK_ / 2; s > 0; s >>= 1) {
    if (d < s) red[d] += red[d + s];
    __syncthreads();
  }
  if (d == 0) qb[bh] = red[0];
}

// ---------------------------------------------------------------------------
// Kernel 1b: qk[b,h,z] = sum_d q[b,h,d] * Wk[h,z,d]
// ---------------------------------------------------------------------------
__global__ void qk_kernel(const float* __restrict__ q,
                          const float* __restrict__ Wk,
                          float* __restrict__ qk) {
  int bh = blockIdx.x;
  int h = bh % H_;
  __shared__ float qs[DK_];
  if (threadIdx.x < DK_) qs[threadIdx.x] = q[bh * DK_ + threadIdx.x];
  __syncthreads();
  for (int z = threadIdx.x; z < Z_; z += blockDim.x) {
    const float* w = Wk + ((size_t)h * Z_ + z) * DK_;
    float acc = 0.f;
#pragma unroll 8
    for (int d = 0; d < DK_; ++d) acc += qs[d] * w[d];
    qk[(size_t)bh * Z_ + z] = acc;
  }
}

// ---------------------------------------------------------------------------
// Kernel 2: scores[b,h,t] = qk[b,h,:]·o_all[b,t,:] + qb[b,h]
// GEMM per batch: M=16(H) x N=T x K=Z, fp32 WMMA 16x16x4.
// Double-buffered async global->LDS staging of o_all tiles.
// ---------------------------------------------------------------------------
#define TT 128   // t-tile per block (8 waves x 16)
#define ZC 32    // z (K) chunk staged in LDS
#define LDP 36   // padded LDS row stride (floats): 144B rows, 16B aligned
#define SLPW 4   // async b128 issues per thread per chunk = TT*ZC/4/256

__global__ __launch_bounds__(256) void scores_kernel(
    const float* __restrict__ o_all, const float* __restrict__ qk,
    const float* __restrict__ qb, float* __restrict__ scores) {
  int b = blockIdx.y;
  int t0 = blockIdx.x * TT;
  int tid = threadIdx.x;
  int lane = tid & 31, wv = tid >> 5;
  int hn = lane & 15;            // A: M (head) / B: N (t) index
  int kb = (lane >> 4) * 2;      // K-pair base within the 4-wide K slab
  __shared__ float lds_o[2 * TT * LDP];

  v8f acc = {};
  const float* ob  = o_all + (size_t)b * T_ * Z_;
  const float* qkb = qk + (size_t)(b * H_) * Z_;

  // stage chunk 0 into buffer 0
#pragma unroll
  for (int i = 0; i < SLPW; ++i) {
    int f = tid + i * 256;
    int row = f >> 3;            // ZC/4 = 8 float4 per row
    int c4  = f & 7;
    cp16_async(ob + (size_t)(t0 + row) * Z_ + c4 * 4, &lds_o[row * LDP + c4 * 4]);
  }

  const int NCH = Z_ / ZC;       // 32 chunks
  for (int c = 0; c < NCH; ++c) {
    float* cur = &lds_o[(c & 1) * (TT * LDP)];
    if (c + 1 < NCH) {
      float* nxt = &lds_o[((c + 1) & 1) * (TT * LDP)];
      int zc1 = (c + 1) * ZC;
#pragma unroll
      for (int i = 0; i < SLPW; ++i) {
        int f = tid + i * 256;
        int row = f >> 3;
        int c4  = f & 7;
        cp16_async(ob + (size_t)(t0 + row) * Z_ + zc1 + c4 * 4,
                   &nxt[row * LDP + c4 * 4]);
      }
      ASYNC_WAIT(SLPW);          // chunk c landed; chunk c+1 still in flight
    } else {
      ASYNC_WAIT(0);
    }
    __syncthreads();

    int zc = c * ZC;
#pragma unroll
    for (int kk = 0; kk < ZC; kk += 4) {
      v2f a  = *(const v2f*)(qkb + (size_t)hn * Z_ + zc + kk + kb);
      v2f bf = *(const v2f*)&cur[(wv * 16 + hn) * LDP + kk + kb];
      acc = __builtin_amdgcn_wmma_f32_16x16x4_f32(false, a, false, bf,
                                                  (short)0, acc, false, false);
    }
    __syncthreads();
  }

  int mb = (lane >> 4) * 8;
  float* srow = scores + (size_t)(b * H_) * T_ + t0 + wv * 16 + hn;
#pragma unroll
  for (int r = 0; r < 8; ++r) {
    int h = mb + r;
    srow[(size_t)h * T_] = acc[r] + qb[b * H_ + h];
  }
}

// ---------------------------------------------------------------------------
// Kernel 3: row softmax over T per (b,h)
// ---------------------------------------------------------------------------
__global__ __launch_bounds__(256) void softmax_kernel(float* __restrict__ s) {
  float* p = s + (size_t)blockIdx.x * T_;
  int tid = threadIdx.x;
  __shared__ float red[256];
  float m = -3.4e38f;
  for (int t = tid; t < T_; t += 256) m = fmaxf(m, p[t]);
  red[tid] = m;
  __syncthreads();
  for (int st = 128; st > 0; st >>= 1) {
    if (tid < st) red[tid] = fmaxf(red[tid], red[tid + st]);
    __syncthreads();
  }
  m = red[0];
  __syncthreads();
  float sum = 0.f;
  for (int t = tid; t < T_; t += 256) {
    float e = __expf(p[t] - m);
    p[t] = e;
    sum += e;
  }
  red[tid] = sum;
  __syncthreads();
  for (int st = 128; st > 0; st >>= 1) {
    if (tid < st) red[tid] += red[tid + st];
    __syncthreads();
  }
  float inv = 1.0f / red[0];
  for (int t = tid; t < T_; t += 256) p[t] *= inv;
}

// ---------------------------------------------------------------------------
// Kernel 4: ctxz[b,h,z] = sum_t s[b,h,t] * o_all[b,t,z]
// GEMM per batch: M=16(H) x N=Z x K=T, fp32 WMMA 16x16x4.
// Double-buffered async global->LDS staging of o_all tiles.
// ---------------------------------------------------------------------------
#define ZT 128    // z-tile (N) per block (8 waves x 16)
#define TC 32     // t (K) chunk staged in LDS
#define LDP2 132  // padded LDS row stride (floats): 528B rows, 16B aligned
#define CLPW 4    // async b128 issues per thread per chunk = TC*ZT/4/256

__global__ __launch_bounds__(256) void ctxz_kernel(
    const float* __restrict__ o_all, const float* __restrict__ s,
    float* __restrict__ ctxz) {
  int b = blockIdx.y;
  int z0 = blockIdx.x * ZT;
  int tid = threadIdx.x;
  int lane = tid & 31, wv = tid >> 5;
  int hn = lane & 15;
  int kb = (lane >> 4) * 2;
  __shared__ float lds_o[2 * TC * LDP2];

  v8f acc = {};
  const float* ob = o_all + (size_t)b * T_ * Z_;
  const float* sb = s + (size_t)(b * H_) * T_;

  // stage chunk 0 into buffer 0
#pragma unroll
  for (int i = 0; i < CLPW; ++i) {
    int f = tid + i * 256;
    int row = f >> 5;            // ZT/4 = 32 float4 per row
    int c4  = f & 31;
    cp16_async(ob + (size_t)row * Z_ + z0 + c4 * 4, &lds_o[row * LDP2 + c4 * 4]);
  }

  const int NCH = T_ / TC;       // 128 chunks
  for (int c = 0; c < NCH; ++c) {
    float* cur = &lds_o[(c & 1) * (TC * LDP2)];
    if (c + 1 < NCH) {
      float* nxt = &lds_o[((c + 1) & 1) * (TC * LDP2)];
      int t1 = (c + 1) * TC;
#pragma unroll
      for (int i = 0; i < CLPW; ++i) {
        int f = tid + i * 256;
        int row = f >> 5;
        int c4  = f & 31;
        cp16_async(ob + (size_t)(t1 + row) * Z_ + z0 + c4 * 4,
                   &nxt[row * LDP2 + c4 * 4]);
      }
      ASYNC_WAIT(CLPW);
    } else {
      ASYNC_WAIT(0);
    }
    __syncthreads();

    int t0 = c * TC;
#pragma unroll
    for (int kk = 0; kk < TC; kk += 4) {
      v2f a = *(const v2f*)(sb + (size_t)hn * T_ + t0 + kk + kb);
      v2f bf;
      bf.x = cur[(kk + kb + 0) * LDP2 + wv * 16 + hn];
      bf.y = cur[(kk + kb + 1) * LDP2 + wv * 16 + hn];
      acc = __builtin_amdgcn_wmma_f32_16x16x4_f32(false, a, false, bf,
                                                  (short)0, acc, false, false);
    }
    __syncthreads();
  }

  int mb = (lane >> 4) * 8;
#pragma unroll
  for (int r = 0; r < 8; ++r) {
    int h = mb + r;
    ctxz[(size_t)(b * H_ + h) * Z_ + z0 + wv * 16 + hn] = acc[r];
  }
}

// ---------------------------------------------------------------------------
// Kernel 5: out[b, h*DK+d] = sum_z ctxz[b,h,z]*Wv[h,z,d] + bv[h,d]
// ---------------------------------------------------------------------------
__global__ void ctx_kernel(const float* __restrict__ ctxz,
                           const float* __restrict__ Wv,
                           const float* __restrict__ bv,
                           float* __restrict__ out) {
  int bh = blockIdx.x;
  int b = bh / H_, h = bh % H_;
  int d = threadIdx.x;
  const float* c = ctxz + (size_t)bh * Z_;
  const float* w = Wv + (size_t)h * Z_ * DK_ + d;
  float acc = bv[h * DK_ + d];
  for (int z = 0; z < Z_; ++z) acc += c[z] * w[(size_t)z * DK_];
  out[(size_t)b * Z_ + h * DK_ + d] = acc;
}

// ---------------------------------------------------------------------------
extern "C" void kernel_launch(void* const* d_in, const int* in_sizes, int n_in,
                              void* d_out, int out_size, void* d_ws, size_t ws_size,
                              hipStream_t stream) {
  const float* o_all  = (const float*)d_in[0];
  const float* o_last = (const float*)d_in[1];
  const float* Wk     = (const float*)d_in[2];
  const float* Wv     = (const float*)d_in[3];
  const float* Wq     = (const float*)d_in[4];
  const float* bk     = (const float*)d_in[5];
  const float* bv     = (const float*)d_in[6];
  const float* bq     = (const float*)d_in[7];
  float* out = (float*)d_out;

  float* ws     = (float*)d_ws;
  float* q      = ws;                              // B*H*DK
  float* qb     = q + B_ * H_ * DK_;               // B*H
  float* qk     = qb + B_ * H_;                    // B*H*Z
  float* scores = qk + B_ * H_ * Z_;               // B*H*T
  float* ctxz   = scores + (size_t)B_ * H_ * T_;   // B*H*Z

  qproj_kernel<<<B_ * H_, DK_, 0, stream>>>(o_last, Wq, bq, bk, q, qb);
  qk_kernel<<<B_ * H_, 256, 0, stream>>>(q, Wk, qk);
  scores_kernel<<<dim3(T_ / TT, B_), 256, 0, stream>>>(o_all, qk, qb, scores);
  softmax_kernel<<<B_ * H_, 256, 0, stream>>>(scores);
  ctxz_kernel<<<dim3(Z_ / ZT, B_), 256, 0, stream>>>(o_all, scores, ctxz);
  ctx_kernel<<<B_ * H_, DK_, 0, stream>>>(ctxz, Wv, bv, out);
}